// Flanger_73160472920642
// MI455X (gfx1250) — compile-verified
//
#include <hip/hip_runtime.h>
#include <math.h>

// ---------------------------------------------------------------------------
// Flanger: out[b,t] = 0.5*x[b,t] + 0.5*w[b,t-d[t]]
//          w[b,t]   = x[b,t] + 0.3*w[b,t-d[t]]        (w=0 for t<0)
//          d[t] = clip(220 + trunc(sin(2*pi*(t*0.5/44100 mod 1))*123), 1, 511)
// d[t] in [97, 343]  -> 96 consecutive timesteps are internally independent.
// One wave32 owns a row-segment: 3 samples/lane/chunk; same-wave LDS ops are
// in-order on CDNA5 (DScnt) -> NO barrier in the sequential chunk chain.
// fb=0.3 decay: 0.3^14 ~= 5e-8, 14*343=4802 <= 4896 warm-up -> segment split.
// Delay schedule packed 3x10-bit per dword, indexed by t/3.
// ---------------------------------------------------------------------------

#define B_ROWS   64
#define BUF_SZ   512
#define BUF_MASK 511
#define WAVE     32
#define CHUNK    96          // = 3 * WAVE, <= min delay (97)
#define SEGMENTS 16          // 64*16 = 1024 single-wave workgroups
#define WARMUP   4896        // 51 full chunks, >= 14 * max_delay(343)
#define FB_C     0.3f
#define MIX_C    0.5f

#define CAP_P    (1 << 17)   // static packed-table capacity (T up to 393216)
__device__ int g_pdly[CAP_P];   // module .bss fallback: no hipMalloc needed

__device__ __forceinline__ int delay_at(int t) {
    // match numpy float64: phase = (t*0.5)/44100 % 1 ; trunc toward zero
    double phase = fmod(((double)t * 0.5) / 44100.0, 1.0);
    double m = sin(6.283185307179586476925286766559 * phase);
    int d = 220 + (int)trunc(m * 123.0);
    d = d < 1 ? 1 : (d > 511 ? 511 : d);
    return d;
}

// one-shot: pack d[3j], d[3j+1], d[3j+2] into 10-bit fields of one dword
__global__ void pack_sched_kernel(int* __restrict__ pd_ws, int Tp, int T,
                                  int useGlobal) {
    int j = blockIdx.x * blockDim.x + threadIdx.x;
    if (j >= Tp) return;
    int* pd = useGlobal ? g_pdly : pd_ws;
    int dv[3];
    #pragma unroll
    for (int k = 0; k < 3; ++k) {
        int t = 3 * j + k;
        dv[k] = (t < T) ? delay_at(t) : 97;
    }
    pd[j] = dv[0] | (dv[1] << 10) | (dv[2] << 20);
}

__global__ __launch_bounds__(WAVE) void flanger_scan_kernel(
        const float* __restrict__ x,
        const int*   __restrict__ pdly,      // nullptr -> use g_pdly
        float*       __restrict__ out,
        int T, int segLen)
{
    __shared__ float buf[BUF_SZ];            // circular w-state, 2 KB
    const int  lane = threadIdx.x;           // 0..31
    const int  s    = blockIdx.x % SEGMENTS;
    const int  b    = blockIdx.x / SEGMENTS;
    const int* __restrict__ pd = pdly ? pdly : g_pdly;

    // zero-init state (same wave -> in-order vs. later ds loads, no barrier)
    {
        float4* b4 = (float4*)buf;
        for (int i = lane; i < BUF_SZ / 4; i += WAVE)
            b4[i] = make_float4(0.f, 0.f, 0.f, 0.f);
    }

    const int outStart = s * segLen;                 // segLen % 96 == 0
    const int tEnd     = min(T, outStart + segLen);
    int t0 = outStart - WARMUP; if (t0 < 0) t0 = 0;  // warm-up = 0 or 51 chunks

    const float* __restrict__ xr   = x   + (size_t)b * T;
    float*       __restrict__ orow = out + (size_t)b * T;

    // full (unguarded) chunk: lane owns t, t+1, t+2 with t = base + 3*lane
    auto chunk_full = [&](int base, bool doStore) {
        const int t = base + 3 * lane;
        // contiguous 12B/lane -> global_load_b96 merges
        float x0 = xr[t], x1 = xr[t + 1], x2 = xr[t + 2];
        // one coalesced b32: base%3==0 -> t/3 = base/3 + lane
        const int p  = pd[base / 3 + lane];
        const int d0 = p & 1023, d1 = (p >> 10) & 1023, d2 = (p >> 20) & 1023;
        // gfx1250 prefetch path (global_prefetch_b8): 4 chunks ahead
        if (base + 4 * CHUNK < tEnd) __builtin_prefetch(xr + t + 4 * CHUNK, 0, 1);
        // reads hit slots [base-343, base-2] mod 512; stride-3 -> no bank conflict
        float w0 = buf[(t     - d0) & BUF_MASK];
        float w1 = buf[(t + 1 - d1) & BUF_MASK];
        float w2 = buf[(t + 2 - d2) & BUF_MASK];
        if (doStore) {                                   // global_store_b96 merges
            orow[t]     = fmaf(w0, MIX_C, x0 * (1.0f - MIX_C));
            orow[t + 1] = fmaf(w1, MIX_C, x1 * (1.0f - MIX_C));
            orow[t + 2] = fmaf(w2, MIX_C, x2 * (1.0f - MIX_C));
        }
        // writes hit slots [base, base+95] mod 512 : disjoint from reads above
        buf[(t)     & BUF_MASK] = fmaf(w0, FB_C, x0);
        buf[(t + 1) & BUF_MASK] = fmaf(w1, FB_C, x1);
        buf[(t + 2) & BUF_MASK] = fmaf(w2, FB_C, x2);
    };

    int base = t0;
    for (; base < outStart; base += CHUNK)               // warm-up (no output)
        chunk_full(base, false);
    for (; base + CHUNK <= tEnd; base += CHUNK)          // main body
        chunk_full(base, true);

    if (base < tEnd) {                                   // global tail only
        const int p = pd[base / 3 + lane];
        #pragma unroll
        for (int k = 0; k < 3; ++k) {
            const int t = base + 3 * lane + k;           // t % 3 == k
            if (t < tEnd) {
                float xv = xr[t];
                int   d  = (p >> (10 * k)) & 1023;
                float w  = buf[(t - d) & BUF_MASK];
                orow[t]  = fmaf(w, MIX_C, xv * (1.0f - MIX_C));
                buf[t & BUF_MASK] = fmaf(w, FB_C, xv);
            }
        }
    }
}

extern "C" void kernel_launch(void* const* d_in, const int* in_sizes, int n_in,
                              void* d_out, int out_size, void* d_ws, size_t ws_size,
                              hipStream_t stream) {
    const float* x   = (const float*)d_in[0];
    float*       out = (float*)d_out;

    const int T  = in_sizes[0] / B_ROWS;                     // 220500
    const int Tp = (T + 2) / 3;                              // packed entries
    // segLen: multiple of CHUNK covering T/SEGMENTS
    const int segLen = CHUNK * ((T + CHUNK * SEGMENTS - 1) / (CHUNK * SEGMENTS));

    const bool useGlobal = (Tp <= CAP_P) ||
                           (ws_size < (size_t)Tp * sizeof(int));
    int* pd_ws = useGlobal ? nullptr : (int*)d_ws;

    pack_sched_kernel<<<(Tp + 255) / 256, 256, 0, stream>>>(
        pd_ws, Tp, T, useGlobal ? 1 : 0);
    flanger_scan_kernel<<<B_ROWS * SEGMENTS, WAVE, 0, stream>>>(
        x, pd_ws, out, T, segLen);
}